// SimpleFinalFusionGate_6536940224997
// MI455X (gfx1250) — compile-verified
//
#include <hip/hip_runtime.h>
#include <math.h>

#define DD 1024
#define THREE_D 3072
#define BB 32
#define TT 2048

typedef __attribute__((ext_vector_type(2))) float v2f;
typedef __attribute__((ext_vector_type(4))) float vf4;
typedef __attribute__((ext_vector_type(8))) float v8f;

// ---------------------------------------------------------------------------
// Kernel 1: LayerNorm of concat(raw_g, spec_g, rel_g) -> g_out [B][3D]
// One block per batch row, 256 threads, each handles 12 strided elements.
// ---------------------------------------------------------------------------
__global__ void ln_kernel(const float* __restrict__ raw_g,
                          const float* __restrict__ spec_g,
                          const float* __restrict__ rel_g,
                          const float* __restrict__ gamma,
                          const float* __restrict__ beta,
                          float* __restrict__ g_out) {
    __shared__ float s_sum[256];
    __shared__ float s_sq[256];
    const int b = blockIdx.x;
    const int tid = threadIdx.x;

    float vals[12];
    float lsum = 0.f, lsq = 0.f;
#pragma unroll
    for (int i = 0; i < 12; ++i) {
        int j = tid + i * 256;          // coalesced
        float x;
        if (j < DD)            x = raw_g [b * DD + j];
        else if (j < 2 * DD)   x = spec_g[b * DD + j - DD];
        else                   x = rel_g [b * DD + j - 2 * DD];
        vals[i] = x;
        lsum += x;
        lsq  += x * x;
    }
    s_sum[tid] = lsum;
    s_sq[tid]  = lsq;
    __syncthreads();
    for (int s = 128; s > 0; s >>= 1) {
        if (tid < s) { s_sum[tid] += s_sum[tid + s]; s_sq[tid] += s_sq[tid + s]; }
        __syncthreads();
    }
    const float mu  = s_sum[0] * (1.0f / THREE_D);
    const float var = s_sq[0] * (1.0f / THREE_D) - mu * mu;
    const float inv = rsqrtf(var + 1e-5f);
#pragma unroll
    for (int i = 0; i < 12; ++i) {
        int j = tid + i * 256;
        g_out[b * THREE_D + j] = (vals[i] - mu) * inv * gamma[j] + beta[j];
    }
}

// ---------------------------------------------------------------------------
// Kernel 2: h = GELU(gLN @ w1 + b1), [32,3072]x[3072,1024] via
// V_WMMA_F32_16X16X4_F32. One wave per 16x16 output tile; 128 tiles total.
// Grid: 16 blocks x 256 threads (8 waves/block).
//
// Per ISA 7.12.2 (32-bit A 16x4): lanes 0-15 carry K={k,k+1} for M=lane,
// lanes 16-31 carry K={k+2,k+3} for M=lane-16.  B 4x16 mirrors the half-wave
// split with N = lane&15.  C/D f32 16x16: VGPR r -> M = r + 8*(lane>>4),
// N = lane&15.
// ---------------------------------------------------------------------------
__global__ void gemm_gelu_kernel(const float* __restrict__ g,    // [B][3D]
                                 const float* __restrict__ w1,   // [3D][D] row-major
                                 const float* __restrict__ b1,   // [D]
                                 float* __restrict__ h)          // [B][D]
{
    const int lane = threadIdx.x & 31;
    const int wave = threadIdx.x >> 5;
    const int tile = blockIdx.x * 8 + wave;   // 0..127
    const int mt   = tile >> 6;               // 0..1   (M tile)
    const int nt   = tile & 63;               // 0..63  (N tile)
    const int half = lane >> 4;               // 0/1
    const int mr   = lane & 15;

    const int m = mt * 16 + mr;               // A row for this lane
    const int n = nt * 16 + mr;               // B/D column for this lane

    const float* __restrict__ ga = g  + m * THREE_D;
    const float* __restrict__ wb = w1 + n;

    v8f acc = {};
#pragma unroll 4
    for (int k = 0; k < THREE_D; k += 4) {
        const int ka = k + 2 * half;
        v2f a, bmat;
        a.x    = ga[ka];
        a.y    = ga[ka + 1];
        bmat.x = wb[ka * DD];
        bmat.y = wb[(ka + 1) * DD];
        acc = __builtin_amdgcn_wmma_f32_16x16x4_f32(
            /*neg_a=*/false, a, /*neg_b=*/false, bmat,
            /*c_mod=*/(short)0, acc, /*reuse_a=*/false, /*reuse_b=*/false);
    }

    const float bias = b1[n];
#pragma unroll
    for (int r = 0; r < 8; ++r) {
        const int row = mt * 16 + r + 8 * half;
        const float x  = acc[r] + bias;
        const float ge = 0.5f * x * (1.0f + erff(x * 0.70710678118654752f)); // exact GELU
        h[row * DD + n] = ge;
    }
}

// ---------------------------------------------------------------------------
// Kernel 3: alpha[b] = sigmoid(h[b,:] . w2 + b2). One block per batch row.
// Writes alpha to workspace (for kernel 4) and to the tail of d_out.
// ---------------------------------------------------------------------------
__global__ void alpha_kernel(const float* __restrict__ h,
                             const float* __restrict__ w2,   // [D]
                             const float* __restrict__ b2,   // [1]
                             float* __restrict__ alpha_ws,
                             float* __restrict__ alpha_out) {
    __shared__ float s[128];
    const int b = blockIdx.x;
    const int tid = threadIdx.x;
    float p = 0.f;
    for (int j = tid; j < DD; j += 128) p += h[b * DD + j] * w2[j];
    s[tid] = p;
    __syncthreads();
    for (int stp = 64; stp > 0; stp >>= 1) {
        if (tid < stp) s[tid] += s[tid + stp];
        __syncthreads();
    }
    if (tid == 0) {
        const float z = s[0] + b2[0];
        const float a = 1.0f / (1.0f + expf(-z));
        alpha_ws[b]  = a;
        alpha_out[b] = a;
    }
}

// ---------------------------------------------------------------------------
// Kernel 4 (bandwidth-dominant, ~805 MB @ 23.3 TB/s ≈ 35 us):
// out = raw + alpha[b] * rel, float4 per thread, non-temporal (streams don't
// fit the 192 MB L2 and are touched once). Each block maps to exactly one
// batch row (524288 float4 per row / 256 = 2048 blocks per row), so alpha[b]
// is a uniform scalar load.
// ---------------------------------------------------------------------------
__global__ void residual_kernel(const float* __restrict__ raw,
                                const float* __restrict__ rel,
                                const float* __restrict__ alpha,
                                float* __restrict__ out) {
    const int bidx = blockIdx.x;
    const int b    = bidx >> 11;                     // 2048 blocks per batch row
    const float a  = alpha[b];
    const long long i = (long long)bidx * 256 + threadIdx.x;

    const vf4* __restrict__ r4 = (const vf4*)raw;
    const vf4* __restrict__ l4 = (const vf4*)rel;
    vf4* __restrict__ o4 = (vf4*)out;

    vf4 x = __builtin_nontemporal_load(&r4[i]);
    vf4 y = __builtin_nontemporal_load(&l4[i]);
    vf4 z;
    z.x = fmaf(a, y.x, x.x);
    z.y = fmaf(a, y.y, x.y);
    z.z = fmaf(a, y.z, x.z);
    z.w = fmaf(a, y.w, x.w);
    __builtin_nontemporal_store(z, &o4[i]);
}

// ---------------------------------------------------------------------------
// Launch
// ---------------------------------------------------------------------------
extern "C" void kernel_launch(void* const* d_in, const int* in_sizes, int n_in,
                              void* d_out, int out_size, void* d_ws, size_t ws_size,
                              hipStream_t stream) {
    (void)in_sizes; (void)n_in; (void)out_size; (void)ws_size;

    const float* raw_residual = (const float*)d_in[0];
    const float* rel_residual = (const float*)d_in[1];
    const float* raw_global   = (const float*)d_in[2];
    const float* spec_global  = (const float*)d_in[3];
    const float* rel_global   = (const float*)d_in[4];
    const float* ln_gamma     = (const float*)d_in[5];
    const float* ln_beta      = (const float*)d_in[6];
    const float* w1           = (const float*)d_in[7];
    const float* b1           = (const float*)d_in[8];
    const float* w2           = (const float*)d_in[9];
    const float* b2           = (const float*)d_in[10];

    float* out_main  = (float*)d_out;
    float* out_alpha = out_main + (size_t)BB * TT * DD;

    // Workspace layout
    char* ws = (char*)d_ws;
    float* ws_g     = (float*)(ws);                                     // 32*3072 f32
    float* ws_h     = (float*)(ws + (size_t)BB * THREE_D * sizeof(float)); // 32*1024 f32
    float* ws_alpha = (float*)(ws + (size_t)BB * THREE_D * sizeof(float)
                                  + (size_t)BB * DD * sizeof(float));   // 32 f32

    ln_kernel<<<BB, 256, 0, stream>>>(raw_global, spec_global, rel_global,
                                      ln_gamma, ln_beta, ws_g);

    gemm_gelu_kernel<<<16, 256, 0, stream>>>(ws_g, w1, b1, ws_h);

    alpha_kernel<<<BB, 128, 0, stream>>>(ws_h, w2, b2, ws_alpha, out_alpha);

    const int total_f4   = BB * TT * DD / 4;     // 16,777,216
    const int grid_resid = total_f4 / 256;       // 65,536 blocks
    residual_kernel<<<grid_resid, 256, 0, stream>>>(raw_residual, rel_residual,
                                                    ws_alpha, out_main);
}